// ComplexWindowAttn_5033701671277
// MI455X (gfx1250) — compile-verified
//
#include <hip/hip_runtime.h>
#include <hip/hip_bf16.h>

// ---------------------------------------------------------------------------
// Complex window attention for MI455X (gfx1250), fp32 WMMA 16x16x4 throughout.
// One workgroup (256 threads = 8 wave32) per window (B = 2048 windows).
// Complex matmuls use the 3-multiply (Karatsuba) form: 25% fewer WMMA ops,
// with the extra VALU adds co-executing against the matrix pipe.
// ---------------------------------------------------------------------------

#define DIMC   96
#define HEADS  6
#define HD     16
#define NTOK   64
#define PSTR   65                    // padded row stride for [96][64] planes
#define PLANE  (DIMC * PSTR)         // 6240 floats per plane
#define SSTR   66                    // scratch stride
#define SLICE_F (16 * SSTR)          // 1056 floats: one 16x64 plane
#define SCR_OFF (8 * PLANE)          // after xr,xi,qr,qi,kr,ki,vr,vi planes
#define SCR_WAVE (2 * SLICE_F)       // re+im scratch per wave
#define REL_OFF (SCR_OFF + 8 * SCR_WAVE)
#define REL_SZ  (HEADS * 225)        // 6 * (2*8-1)^2
#define LDS_FLOATS (REL_OFF + REL_SZ)   // 68166 floats = 272,664 bytes

typedef __attribute__((ext_vector_type(2))) float v2f;
typedef __attribute__((ext_vector_type(8))) float v8f;

__device__ __forceinline__ v8f wmma4(v2f a, v2f b, v8f c) {
  // D = A(16x4) * B(4x16) + C(16x16), fp32.
  return __builtin_amdgcn_wmma_f32_16x16x4_f32(
      /*neg_a=*/false, a, /*neg_b=*/false, b,
      /*c_mod=*/(short)0, c, /*reuse_a=*/false, /*reuse_b=*/false);
}

#define Z8 {0.f,0.f,0.f,0.f,0.f,0.f,0.f,0.f}

__global__ __launch_bounds__(256, 1)
void cwin_attn_kernel(const float* __restrict__ x_real,
                      const float* __restrict__ x_imag,
                      const float* __restrict__ qkv_wr,
                      const float* __restrict__ qkv_wi,
                      const float* __restrict__ proj_wr,
                      const float* __restrict__ proj_wi,
                      const float* __restrict__ rel,
                      float* __restrict__ out)
{
  __shared__ float lds[LDS_FLOATS];

  const int tid  = threadIdx.x;
  const int wave = tid >> 5;
  const int lane = tid & 31;
  const int lh   = lane >> 4;     // 16-lane half (selects K pair / M+8)
  const int ln   = lane & 15;     // lane-in-half
  const size_t b = blockIdx.x;

  // ---------------- Stage A: stage x and rel into LDS --------------------
  {
    const float* xr = x_real + b * (DIMC * NTOK);
    const float* xi = x_imag + b * (DIMC * NTOK);
    for (int i = tid; i < DIMC * NTOK; i += 256) {
      const int c = i >> 6, n = i & 63;
      lds[c * PSTR + n]         = xr[i];
      lds[PLANE + c * PSTR + n] = xi[i];
    }
    for (int i = tid; i < REL_SZ; i += 256) lds[REL_OFF + i] = rel[i];
  }
  __syncthreads();

  // ---------------- Stage B: qkv = Wqkv (288x96 cplx) * x (96x64 cplx) ----
  // 18 M-tiles x 4 N-tiles = 72 output tiles over 8 waves.
  // Karatsuba: P1=Wr*Xr, P2=Wi*Xi, P3=(Wr+Wi)(Xr+Xi); re=P1-P2, im=P3-P1-P2.
  for (int tile = wave; tile < 72; tile += 8) {
    const int mt = tile >> 2;           // 0..17
    const int nt = tile & 3;            // 0..3
    const int arow = mt * 16 + ln;      // weight row (A matrix M)
    const int bcol = nt * 16 + ln;      // token column (B matrix N)
    v8f p1 = Z8, p2 = Z8, p3 = Z8;
    for (int k0 = 0; k0 < DIMC; k0 += 4) {
      const int kk = k0 + 2 * lh;
      const v2f wr = *(const v2f*)(qkv_wr + arow * DIMC + kk);
      const v2f wi = *(const v2f*)(qkv_wi + arow * DIMC + kk);
      const v2f ws = wr + wi;
      v2f br, bi;
      br.x = lds[kk * PSTR + bcol];            br.y = lds[(kk + 1) * PSTR + bcol];
      bi.x = lds[PLANE + kk * PSTR + bcol];    bi.y = lds[PLANE + (kk + 1) * PSTR + bcol];
      const v2f bs = br + bi;
      p1 = wmma4(wr, br, p1);
      p2 = wmma4(wi, bi, p2);
      p3 = wmma4(ws, bs, p3);
    }
    const v8f cre = p1 - p2;
    const v8f cim = p3 - p1 - p2;
#pragma unroll
    for (int r = 0; r < 8; ++r) {
      const int o = mt * 16 + r + 8 * lh;      // 0..287
      const int p = o / 96;                    // 0=q 1=k 2=v
      const int c = o - p * 96;
      lds[(2 + 2 * p) * PLANE + c * PSTR + bcol] = cre[r];
      lds[(3 + 2 * p) * PLANE + c * PSTR + bcol] = cim[r];
    }
  }
  __syncthreads();

  // ---------------- Stage D: attention, one 16x64 slice per wave-task -----
  // 6 heads * 4 row-blocks = 24 slices over 8 waves (3 rounds).
  for (int slice = wave; slice < 24; slice += 8) {
    const int h  = slice >> 2;
    const int rb = slice & 3;
    const int hb = h * 16;

    // attn = q * conj(k)^T, Karatsuba with conj folded in:
    // P1=qr*kr, P2=qi*ki, P3=(qr+qi)(kr-ki); re=P1+P2, im=P3-P1+P2.
    v8f p1[4] = {Z8, Z8, Z8, Z8};
    v8f p2[4] = {Z8, Z8, Z8, Z8};
    v8f p3[4] = {Z8, Z8, Z8, Z8};
    const int qcol = rb * 16 + ln;
    for (int k0 = 0; k0 < HD; k0 += 4) {
      const int kk = k0 + 2 * lh;
      v2f qr, qi;
      qr.x = lds[2 * PLANE + (hb + kk) * PSTR + qcol];
      qr.y = lds[2 * PLANE + (hb + kk + 1) * PSTR + qcol];
      qi.x = lds[3 * PLANE + (hb + kk) * PSTR + qcol];
      qi.y = lds[3 * PLANE + (hb + kk + 1) * PSTR + qcol];
      const v2f qs = qr + qi;
#pragma unroll
      for (int t = 0; t < 4; ++t) {
        const int kcol = t * 16 + ln;
        v2f kr, ki;
        kr.x = lds[4 * PLANE + (hb + kk) * PSTR + kcol];
        kr.y = lds[4 * PLANE + (hb + kk + 1) * PSTR + kcol];
        ki.x = lds[5 * PLANE + (hb + kk) * PSTR + kcol];
        ki.y = lds[5 * PLANE + (hb + kk + 1) * PSTR + kcol];
        const v2f kd = kr - ki;
        p1[t] = wmma4(qr, kr, p1[t]);
        p2[t] = wmma4(qi, ki, p2[t]);
        p3[t] = wmma4(qs, kd, p3[t]);
      }
    }
    v8f are[4], aim[4];
#pragma unroll
    for (int t = 0; t < 4; ++t) {
      are[t] = p1[t] + p2[t];
      aim[t] = p3[t] - p1[t] + p2[t];
    }

    // scale + relative-position bias + magnitude
    v8f mag[4];
#pragma unroll
    for (int t = 0; t < 4; ++t) {
      const int mm = t * 16 + ln;
#pragma unroll
      for (int r = 0; r < 8; ++r) {
        const int nn  = rb * 16 + r + 8 * lh;
        const int idx = ((nn >> 3) - (mm >> 3) + 7) * 15 + ((nn & 7) - (mm & 7) + 7);
        const float re = are[t][r] * 0.25f + lds[REL_OFF + h * 225 + idx];
        const float im = aim[t][r] * 0.25f;
        are[t][r] = re;
        aim[t][r] = im;
        mag[t][r] = sqrtf(re * re + im * im);
      }
    }

    // magnitude softmax along m (row lives in one 16-lane half x 4 tiles)
#pragma unroll
    for (int r = 0; r < 8; ++r) {
      float mx = fmaxf(fmaxf(mag[0][r], mag[1][r]), fmaxf(mag[2][r], mag[3][r]));
#pragma unroll
      for (int off = 8; off >= 1; off >>= 1) mx = fmaxf(mx, __shfl_xor(mx, off, 16));
      float e[4], s = 0.f;
#pragma unroll
      for (int t = 0; t < 4; ++t) { e[t] = __expf(mag[t][r] - mx); s += e[t]; }
#pragma unroll
      for (int off = 8; off >= 1; off >>= 1) s += __shfl_xor(s, off, 16);
      const float inv = 1.0f / s;
#pragma unroll
      for (int t = 0; t < 4; ++t) {
        const float f = e[t] * inv / (mag[t][r] + 1e-8f);
        are[t][r] *= f;
        aim[t][r] *= f;
      }
    }

    // spill slice to per-wave scratch so it can be re-read in A-operand layout
    const int sb = SCR_OFF + wave * SCR_WAVE;
#pragma unroll
    for (int t = 0; t < 4; ++t) {
#pragma unroll
      for (int r = 0; r < 8; ++r) {
        const int row = r + 8 * lh;
        lds[sb + row * SSTR + t * 16 + ln]           = are[t][r];
        lds[sb + SLICE_F + row * SSTR + t * 16 + ln] = aim[t][r];
      }
    }

    // out = attn * v, Karatsuba: P1=ar*vr, P2=ai*vi, P3=(ar+ai)(vr+vi)
    v8f q1 = Z8, q2 = Z8, q3 = Z8;
    for (int k0 = 0; k0 < NTOK; k0 += 4) {
      const int kk = k0 + 2 * lh;
      v2f ar, ai, vr, vi;
      ar.x = lds[sb + ln * SSTR + kk];           ar.y = lds[sb + ln * SSTR + kk + 1];
      ai.x = lds[sb + SLICE_F + ln * SSTR + kk]; ai.y = lds[sb + SLICE_F + ln * SSTR + kk + 1];
      vr.x = lds[6 * PLANE + (hb + ln) * PSTR + kk];
      vr.y = lds[6 * PLANE + (hb + ln) * PSTR + kk + 1];
      vi.x = lds[7 * PLANE + (hb + ln) * PSTR + kk];
      vi.y = lds[7 * PLANE + (hb + ln) * PSTR + kk + 1];
      const v2f as = ar + ai;
      const v2f vs = vr + vi;
      q1 = wmma4(ar, vr, q1);
      q2 = wmma4(ai, vi, q2);
      q3 = wmma4(as, vs, q3);
    }
    const v8f ore = q1 - q2;
    const v8f oim = q3 - q1 - q2;

    // write out slice into channel-major out planes (reuse x region)
#pragma unroll
    for (int r = 0; r < 8; ++r) {
      const int nn = rb * 16 + r + 8 * lh;
      lds[(hb + ln) * PSTR + nn]         = ore[r];   // plane 0: out_re[c][n]
      lds[PLANE + (hb + ln) * PSTR + nn] = oim[r];   // plane 1: out_im[c][n]
    }
  }
  __syncthreads();

  // ---------------- Stage E: proj (96x96 cplx) * out (96x64 cplx) ---------
  // 6 M-tiles x 4 N-tiles = 24 tiles over 8 waves.
  for (int tile = wave; tile < 24; tile += 8) {
    const int mt = tile >> 2;
    const int nt = tile & 3;
    const int arow = mt * 16 + ln;
    const int bcol = nt * 16 + ln;
    v8f p1 = Z8, p2 = Z8, p3 = Z8;
    for (int k0 = 0; k0 < DIMC; k0 += 4) {
      const int kk = k0 + 2 * lh;
      const v2f wr = *(const v2f*)(proj_wr + arow * DIMC + kk);
      const v2f wi = *(const v2f*)(proj_wi + arow * DIMC + kk);
      const v2f ws = wr + wi;
      v2f br, bi;
      br.x = lds[kk * PSTR + bcol];         br.y = lds[(kk + 1) * PSTR + bcol];
      bi.x = lds[PLANE + kk * PSTR + bcol]; bi.y = lds[PLANE + (kk + 1) * PSTR + bcol];
      const v2f bs = br + bi;
      p1 = wmma4(wr, br, p1);
      p2 = wmma4(wi, bi, p2);
      p3 = wmma4(ws, bs, p3);
    }
    const v8f cre = p1 - p2;
    const v8f cim = p3 - p1 - p2;
#pragma unroll
    for (int r = 0; r < 8; ++r) {
      const int o = mt * 16 + r + 8 * lh;
      const int n = nt * 16 + ln;
      float2 val = make_float2(cre[r], cim[r]);   // complex64 interleaved
      *(float2*)(out + ((b * DIMC + o) * NTOK + n) * 2) = val;
    }
  }
}

extern "C" void kernel_launch(void* const* d_in, const int* in_sizes, int n_in,
                              void* d_out, int out_size, void* d_ws, size_t ws_size,
                              hipStream_t stream) {
  const float* x_real  = (const float*)d_in[0];
  const float* x_imag  = (const float*)d_in[1];
  const float* qkv_wr  = (const float*)d_in[2];
  const float* qkv_wi  = (const float*)d_in[3];
  const float* proj_wr = (const float*)d_in[4];
  const float* proj_wi = (const float*)d_in[5];
  const float* rel     = (const float*)d_in[6];
  float* out = (float*)d_out;

  const int B = in_sizes[0] / (DIMC * NTOK);   // 2048
  cwin_attn_kernel<<<B, 256, 0, stream>>>(x_real, x_imag, qkv_wr, qkv_wi,
                                          proj_wr, proj_wi, rel, out);
}